// VectorQuantizer_24094766531103
// MI455X (gfx1250) — compile-verified
//
#include <hip/hip_runtime.h>
#include <hip/hip_bf16.h>

typedef __attribute__((ext_vector_type(2))) float v2f;
typedef __attribute__((ext_vector_type(8))) float v8f;

#define VQ_NUM_EMB 8192
#define VQ_DIM 4
#define VQ_NROWS 32768                 // B*L = 8*4096
#define VQ_NTILE_N (VQ_NUM_EMB / 16)   // 512 N-tiles
#define WAVES_PER_BLOCK 8
#define UNROLL 4

__device__ __forceinline__ void vq_argmax_update(const v8f& c, int n,
                                                 float best[8], int bidx[8]) {
#pragma unroll
    for (int r = 0; r < 8; ++r) {
        if (c[r] > best[r]) { best[r] = c[r]; bidx[r] = n; }
    }
}

// One wave handles one 16-row tile of x. K=4 == full depth, so each 16x16
// similarity tile is exactly one v_wmma_f32_16x16x4_f32.
__global__ __launch_bounds__(256) void VectorQuantizer_vq_loss_kernel(
    const float* __restrict__ x, const float* __restrict__ emb,
    float* __restrict__ out)
{
    const int lane  = threadIdx.x & 31;
    const int wave  = threadIdx.x >> 5;
    const int tileM = blockIdx.x * WAVES_PER_BLOCK + wave;   // 0..2047
    const int kHalf = lane >> 4;   // which K-pair this half-wave holds
    const int nl    = lane & 15;   // M (for A) / N (for B,C) within tile

    // A tile (16x4): lane L holds x[tileM*16 + L%16][K=(L/16)*2 + j]
    const int mrow = tileM * 16 + nl;
    const v2f a = *(const v2f*)(x + (unsigned)(mrow * VQ_DIM + kHalf * 2));

    // Per-lane element offset of this lane's B slot inside an N-tile
    // (16 embeddings * 4 floats = 64 elements = 256 bytes per tile).
    const unsigned laneElem = (unsigned)(nl * VQ_DIM + kHalf * 2);

    float best[8];
    int   bidx[8];
#pragma unroll
    for (int r = 0; r < 8; ++r) { best[r] = -3.402823466e+38f; bidx[r] = 0; }

#pragma unroll 1
    for (int t = 0; t < VQ_NTILE_N; t += UNROLL) {
        const unsigned e0 = laneElem + (unsigned)t * 64u;
        // 4 independent B-tile loads: sgpr base + vgpr offset + imm offsets
        const v2f b0 = *(const v2f*)(emb + e0);
        const v2f b1 = *(const v2f*)(emb + e0 + 64u);
        const v2f b2 = *(const v2f*)(emb + e0 + 128u);
        const v2f b3 = *(const v2f*)(emb + e0 + 192u);

        // Prefetch next group (wrapped so the loop body stays uniform).
        const unsigned tn = (unsigned)((t + UNROLL) & (VQ_NTILE_N - 1));
        __builtin_prefetch(emb + laneElem + tn * 64u, 0, 3);

        v8f c0 = {}, c1 = {}, c2 = {}, c3 = {};
        c0 = __builtin_amdgcn_wmma_f32_16x16x4_f32(false, a, false, b0,
                                                   (short)0, c0, false, false);
        c1 = __builtin_amdgcn_wmma_f32_16x16x4_f32(false, a, false, b1,
                                                   (short)0, c1, false, false);
        c2 = __builtin_amdgcn_wmma_f32_16x16x4_f32(false, a, false, b2,
                                                   (short)0, c2, false, false);
        c3 = __builtin_amdgcn_wmma_f32_16x16x4_f32(false, a, false, b3,
                                                   (short)0, c3, false, false);

        const int nbase = t * 16 + nl;
        vq_argmax_update(c0, nbase,      best, bidx);
        vq_argmax_update(c1, nbase + 16, best, bidx);
        vq_argmax_update(c2, nbase + 32, best, bidx);
        vq_argmax_update(c3, nbase + 48, best, bidx);
    }

    // Argmax reduce over the 16 lanes of each half-wave (same rows, disjoint N).
#pragma unroll
    for (int mask = 1; mask < 16; mask <<= 1) {
#pragma unroll
        for (int r = 0; r < 8; ++r) {
            float ov = __shfl_xor(best[r], mask, 32);
            int   oi = __shfl_xor(bidx[r], mask, 32);
            if (ov > best[r] || (ov == best[r] && oi < bidx[r])) {
                best[r] = ov; bidx[r] = oi;
            }
        }
    }

    // L1 accumulation: lane 0 covers rows 0..7, lane 16 covers rows 8..15.
    float s = 0.0f;
    if (nl == 0) {
#pragma unroll
        for (int r = 0; r < 8; ++r) {
            const int m = tileM * 16 + kHalf * 8 + r;
            const float4 xr = ((const float4*)x)[m];
            const float4 er = ((const float4*)emb)[bidx[r]];
            s += fabsf(xr.x - er.x) + fabsf(xr.y - er.y) +
                 fabsf(xr.z - er.z) + fabsf(xr.w - er.w);
        }
    }
#pragma unroll
    for (int off = 16; off > 0; off >>= 1) s += __shfl_xor(s, off, 32);

    // loss = reg + embedding = 2 * mean|x - e|, mean over B*L*DIM = 131072 elems
    if (lane == 0) atomicAdd(out, s * (2.0f / 131072.0f));
}

extern "C" void kernel_launch(void* const* d_in, const int* in_sizes, int n_in,
                              void* d_out, int out_size, void* d_ws, size_t ws_size,
                              hipStream_t stream) {
    const float* x   = (const float*)d_in[0];        // [8,4096,4]  f32
    const float* emb = (const float*)d_in[1];        // [8192,4]    f32
    float* out = (float*)d_out;                      // scalar f32

    hipMemsetAsync(out, 0, sizeof(float), stream);   // graph-capture-safe memset node

    const int numTilesM = VQ_NROWS / 16;             // 2048 waves
    dim3 grid(numTilesM / WAVES_PER_BLOCK);          // 256 blocks
    dim3 block(WAVES_PER_BLOCK * 32);                // 256 threads = 8 wave32
    VectorQuantizer_vq_loss_kernel<<<grid, block, 0, stream>>>(x, emb, out);
}